// EnhancedSupConLoss_9560597201221
// MI455X (gfx1250) — compile-verified
//
#include <hip/hip_runtime.h>
#include <hip/hip_bf16.h>
#include <math.h>

// ---------------------------------------------------------------------------
// EnhancedSupConLoss for MI455X (gfx1250, wave32).
//   features: [2048, 2, 256] f32  -> flatten to N=4096 rows, D=256
//   labels:   [2048] int          -> lab[j] = labels[j>>1] (repeat_interleave)
// Pipeline:
//   K1 normalize rows (also writes transpose for coalesced WMMA B operand)
//   K2 Gram matrix S = (fN @ fN^T)/TEMP via V_WMMA_F32_16X16X4_F32,
//      32x32 output per wave (2x2 tiles) for 2x lower L2 traffic
//   K3 per-row: async-stream row into LDS (ASYNCcnt path), rowmax, margin,
//      exact k-th-largest radix select, tie-exact top-k exp sum, row loss
//   K4 deterministic tree-mean -> d_out[0]
// ---------------------------------------------------------------------------

#define NROW 4096
#define DDIM 256
#define KSEL 1433          // int(4096 * 0.35)
#define MARGIN 0.2f
#define INV_TEMP 20.0f     // 1 / 0.05
#define T_OVER_BT (0.05f / 0.07f)

typedef __attribute__((ext_vector_type(2))) float v2f;
typedef __attribute__((ext_vector_type(8))) float v8f;

// ---------------------------- K1: normalize --------------------------------
__global__ void __launch_bounds__(256)
normalize_kernel(const float* __restrict__ f,
                 float* __restrict__ fN,
                 float* __restrict__ fT) {
    __shared__ float red[256];
    const int row = blockIdx.x;    // 0..4095
    const int t   = threadIdx.x;   // 0..255
    const float v = f[(size_t)row * DDIM + t];
    red[t] = v * v;
    __syncthreads();
    for (int s = 128; s > 0; s >>= 1) {
        if (t < s) red[t] += red[t + s];
        __syncthreads();
    }
    const float inv = 1.0f / fmaxf(sqrtf(red[0]), 1e-12f);
    const float nv = v * inv;
    fN[(size_t)row * DDIM + t] = nv;
    fT[(size_t)t * NROW + row] = nv;   // transpose for B operand
}

// ----------------------- K2: Gram matrix via WMMA --------------------------
// One wave computes a 32x32 tile (2x2 of 16x16) so each A/B fragment feeds
// two WMMAs: 64 B of L2 traffic per output instead of 128 B.
// A-frag (16x4, MxK): lanes 0-15 hold {K=k,k+1} of row M=lane; lanes 16-31
// hold {K=k+2,k+3} of row M=lane-16.  B-frag (4x16, KxN): VGPR0 lanes 0-15 =
// row K=k, lanes 16-31 = row K=k+1; VGPR1 = rows k+2 / k+3.
__device__ __forceinline__ v8f wmma4(v2f a, v2f b, v8f c) {
    // (neg_a, A, neg_b, B, c_mod, C, reuse_a, reuse_b)
    return __builtin_amdgcn_wmma_f32_16x16x4_f32(
        false, a, false, b, (short)0, c, false, false);
}

__device__ __forceinline__ void store_tile(float* __restrict__ S, v8f acc,
                                           int row0, int col) {
    #pragma unroll
    for (int r = 0; r < 8; ++r)
        S[(size_t)(row0 + r) * NROW + col] = acc[r] * INV_TEMP;
}

__global__ void __launch_bounds__(32)
gram_wmma_kernel(const float* __restrict__ fN,
                 const float* __restrict__ fT,
                 float* __restrict__ S) {
    const int tm   = blockIdx.x;         // 32-row tile (0..127)
    const int tn   = blockIdx.y;         // 32-col tile (0..127)
    const int lane = threadIdx.x;        // 0..31
    const int half = lane >> 4;          // 0|1
    const int l16  = lane & 15;

    const float* a0p = fN + (size_t)(tm * 32 + l16) * DDIM + half * 2;
    const float* a1p = a0p + (size_t)16 * DDIM;
    const float* b0p = fT + (size_t)half * NROW + tn * 32 + l16;
    const float* b1p = b0p + 16;

    v8f acc00 = {}, acc01 = {}, acc10 = {}, acc11 = {};
    #pragma unroll 4
    for (int k = 0; k < DDIM; k += 4) {
        __builtin_prefetch(a0p + k + 64, 0, 3);   // global_prefetch_b8, near
        v2f a0, a1, b0, b1;
        a0.x = a0p[k];  a0.y = a0p[k + 1];
        a1.x = a1p[k];  a1.y = a1p[k + 1];
        b0.x = b0p[(size_t)k * NROW];  b0.y = b0p[(size_t)(k + 2) * NROW];
        b1.x = b1p[(size_t)k * NROW];  b1.y = b1p[(size_t)(k + 2) * NROW];
        acc00 = wmma4(a0, b0, acc00);
        acc01 = wmma4(a0, b1, acc01);
        acc10 = wmma4(a1, b0, acc10);
        acc11 = wmma4(a1, b1, acc11);
    }

    // C/D layout: VGPR r, lanes 0-15 -> M=r, lanes 16-31 -> M=r+8.
    const int r0 = tm * 32 + half * 8;
    const int c0 = tn * 32 + l16;
    store_tile(S, acc00, r0,      c0);
    store_tile(S, acc01, r0,      c0 + 16);
    store_tile(S, acc10, r0 + 16, c0);
    store_tile(S, acc11, r0 + 16, c0 + 16);
}

// ----------------------- K3: per-row loss ----------------------------------
__device__ __forceinline__ float blkSumF(float v, float* sred, int t) {
    sred[t] = v; __syncthreads();
    for (int s = 128; s > 0; s >>= 1) { if (t < s) sred[t] += sred[t + s]; __syncthreads(); }
    float r = sred[0]; __syncthreads();
    return r;
}
__device__ __forceinline__ int blkSumI(int v, int* ired, int t) {
    ired[t] = v; __syncthreads();
    for (int s = 128; s > 0; s >>= 1) { if (t < s) ired[t] += ired[t + s]; __syncthreads(); }
    int r = ired[0]; __syncthreads();
    return r;
}

__global__ void __launch_bounds__(256)
row_loss_kernel(const float* __restrict__ S,
                const int* __restrict__ labels,
                float* __restrict__ rowLoss) {
    __shared__ unsigned buf[NROW];    // 16KB: row values, then in-place keys
    __shared__ unsigned hist[256];
    __shared__ float    sred[256];
    __shared__ int      ired[256];
    __shared__ unsigned sPrefix;
    __shared__ int      sKRem;

    const int i = blockIdx.x;
    const int t = threadIdx.x;
    const int labI = labels[i >> 1];
    const float* rowG = S + (size_t)i * NROW;

    // Stream the 16KB row into LDS once via gfx1250 async load-to-LDS
    // (ASYNCcnt path, no VGPR round-trip). Flat shared addr truncated to
    // 32 bits is the wave-relative LDS byte offset (ISA 10.2 aperture rule).
    {
        const unsigned ldsBase = (unsigned)(size_t)(&buf[0]);
        #pragma unroll
        for (int it = 0; it < 4; ++it) {
            const int e = (it * 256 + t) * 4;            // float index, 16B step
            const unsigned lAddr = ldsBase + (unsigned)(e * 4);
            const float* g = rowG + e;
            asm volatile("global_load_async_to_lds_b128 %0, %1, off"
                         :: "v"(lAddr), "v"(g) : "memory");
        }
        asm volatile("s_wait_asynccnt 0x0" ::: "memory");
        __syncthreads();
    }

    // pass A: row max (pre-margin, matching reference) — from LDS
    float mx = -3.4e38f;
    for (int j = t; j < NROW; j += 256) mx = fmaxf(mx, __uint_as_float(buf[j]));
    sred[t] = mx; __syncthreads();
    for (int s = 128; s > 0; s >>= 1) { if (t < s) sred[t] = fmaxf(sred[t], sred[t + s]); __syncthreads(); }
    const float rowMax = sred[0];
    __syncthreads();

    // pass B: adc = s - rowMax (- margin on negatives); positive stats;
    // overwrite LDS in place with monotone keys. All negative adc < 0
    // strictly => key = ~bits ascends with the float, in (0, 0x7fffffff];
    // positives get key 0 (excluded from selection).
    float posSumAdc = 0.0f, posSumExp = 0.0f;
    int posCnt = 0, negCnt = 0;
    for (int j = t; j < NROW; j += 256) {
        const float s  = __uint_as_float(buf[j]);
        const bool pos = (labels[j >> 1] == labI);
        const float adc = s - rowMax - (pos ? 0.0f : MARGIN);
        if (pos) {
            posSumAdc += adc;
            posSumExp += expf(adc);
            ++posCnt;
            buf[j] = 0u;
        } else {
            ++negCnt;
            buf[j] = ~__float_as_uint(adc);
        }
    }
    const float posSumAdcT = blkSumF(posSumAdc, sred, t);
    const float posSumExpT = blkSumF(posSumExp, sred, t);
    const int   posCntT    = blkSumI(posCnt, ired, t);
    const int   negCntT    = blkSumI(negCnt, ired, t);

    const int kk = (negCntT < KSEL) ? negCntT : KSEL;
    unsigned vk = 0u;  // 0 => "take all negatives" (keys > 0)

    if (negCntT > kk) {
        // exact k-th largest key via 4x 8-bit MSB-first radix select
        unsigned prefix = 0u, prefMask = 0u;
        int kRem = kk;
        for (int pass = 0; pass < 4; ++pass) {
            const int shift = 24 - 8 * pass;
            hist[t] = 0u; __syncthreads();
            for (int j = t; j < NROW; j += 256) {
                const unsigned key = buf[j];
                if (key && (key & prefMask) == prefix)
                    atomicAdd(&hist[(key >> shift) & 255u], 1u);
            }
            __syncthreads();
            if (t == 0) {
                int acc = 0, b = 255;
                for (; b >= 0; --b) {
                    acc += (int)hist[b];
                    if (acc >= kRem) break;
                }
                sPrefix = prefix | ((unsigned)b << shift);
                sKRem   = kRem - (acc - (int)hist[b]);
            }
            __syncthreads();
            prefix   = sPrefix;
            kRem     = sKRem;
            prefMask |= (0xFFu << shift);
            __syncthreads();
        }
        vk = prefix;
    }

    // pass C: tie-exact top-k exp sum: strictly-greater part + m copies of vk
    float sumG = 0.0f; int cntG = 0;
    for (int j = t; j < NROW; j += 256) {
        const unsigned key = buf[j];
        if (key > vk) {
            sumG += expf(__uint_as_float(~key));
            ++cntG;
        }
    }
    const float sumGT = blkSumF(sumG, sred, t);
    const int   cntGT = blkSumI(cntG, ired, t);

    if (t == 0) {
        float topSum = sumGT;
        const int m = kk - cntGT;
        if (vk != 0u && m > 0)
            topSum += (float)m * expf(__uint_as_float(~vk));
        const float denom = posSumExpT + topSum + 1e-12f;
        const float lg = logf(denom);
        const float meanLP = (posSumAdcT - (float)posCntT * lg) /
                             ((float)posCntT + 1e-12f);
        rowLoss[i] = -T_OVER_BT * meanLP;
    }
}

// ----------------------- K4: deterministic mean ----------------------------
__global__ void __launch_bounds__(256)
mean_kernel(const float* __restrict__ rowLoss, float* __restrict__ out) {
    __shared__ float sred[256];
    const int t = threadIdx.x;
    float s = 0.0f;
    for (int j = t; j < NROW; j += 256) s += rowLoss[j];  // fixed order
    sred[t] = s; __syncthreads();
    for (int st = 128; st > 0; st >>= 1) { if (t < st) sred[t] += sred[t + st]; __syncthreads(); }
    if (t == 0) out[0] = sred[0] * (1.0f / (float)NROW);
}

// ---------------------------------------------------------------------------
extern "C" void kernel_launch(void* const* d_in, const int* in_sizes, int n_in,
                              void* d_out, int out_size, void* d_ws, size_t ws_size,
                              hipStream_t stream) {
    const float* feats  = (const float*)d_in[0];   // [2048,2,256] f32
    const int*   labels = (const int*)d_in[1];     // [2048] int
    float*       out    = (float*)d_out;           // scalar loss

    char* ws = (char*)d_ws;
    float* fN      = (float*)(ws);                                   //  4 MB
    float* fT      = (float*)(ws + ((size_t)4  << 20));              //  4 MB
    float* S       = (float*)(ws + ((size_t)8  << 20));              // 64 MB
    float* rowLoss = (float*)(ws + ((size_t)8  << 20)
                                 + (size_t)NROW * NROW * sizeof(float));

    normalize_kernel<<<NROW, 256, 0, stream>>>(feats, fN, fT);

    dim3 gGram(NROW / 32, NROW / 32);   // 128 x 128 tiles of 32x32, 1 wave each
    gram_wmma_kernel<<<gGram, 32, 0, stream>>>(fN, fT, S);

    row_loss_kernel<<<NROW, 256, 0, stream>>>(S, labels, rowLoss);

    mean_kernel<<<1, 256, 0, stream>>>(rowLoss, out);
}